// PixelFeatureExchange_53687091200545
// MI455X (gfx1250) — compile-verified
//
#include <hip/hip_runtime.h>

// PixelFeatureExchange: out[d][b,c,i,j] = x[(d + i%3 + j%3) % 3][b,c,i,j]
// Pure streaming permutation: 768 MiB total traffic, ~34 us floor @ 23.3 TB/s.
// Strategy: one thread loads one float4 from each of the 3 inputs (48 B) and
// produces the corresponding float4 of ALL THREE outputs (48 B) -> minimal
// traffic, perfectly coalesced 128-bit accesses, NT cache hints (single-use
// stream, don't thrash the 192 MB L2), plus gfx1250 global_prefetch one
// grid-stride iteration ahead.

typedef float f4 __attribute__((ext_vector_type(4)));

__global__ __launch_bounds__(256) void pfe_kernel(
    const f4* __restrict__ x0, const f4* __restrict__ x1,
    const f4* __restrict__ x2, f4* __restrict__ out, unsigned n4) {
  const unsigned stride = gridDim.x * blockDim.x;
  for (unsigned idx = blockIdx.x * blockDim.x + threadIdx.x; idx < n4;
       idx += stride) {
    // gfx1250 global_prefetch_b8: pull next iteration's lines toward L2.
    if (idx + stride < n4) {
      __builtin_prefetch((const void*)(x0 + idx + stride), 0, 0);
      __builtin_prefetch((const void*)(x1 + idx + stride), 0, 0);
      __builtin_prefetch((const void*)(x2 + idx + stride), 0, 0);
    }

    // Element offset within one [B,C,H,W] tensor; W == H == 256.
    const unsigned e = idx << 2;        // multiple of 4, never crosses a row
    const unsigned j = e & 255u;        // column
    const unsigned i = (e >> 8) & 255u; // row
    const unsigned t0 = (i + j) % 3u;   // == (i%3 + j%3) % 3

    // 3 x 16B coalesced non-temporal loads (each input byte used once).
    const f4 a = __builtin_nontemporal_load(x0 + idx);
    const f4 b = __builtin_nontemporal_load(x1 + idx);
    const f4 c = __builtin_nontemporal_load(x2 + idx);

    // cand[m].comp_k = input[(m+k)%3].comp_k  (note k=3 == k=0 mod 3)
    const f4 cand0 = (f4){a.x, b.y, c.z, a.w};
    const f4 cand1 = (f4){b.x, c.y, a.z, b.w};
    const f4 cand2 = (f4){c.x, a.y, b.z, c.w};

    // out[d] = cand[(t0 + d) % 3] -- per-lane v_cndmask selection.
    const f4 o0 = (t0 == 0u) ? cand0 : ((t0 == 1u) ? cand1 : cand2);
    const f4 o1 = (t0 == 0u) ? cand1 : ((t0 == 1u) ? cand2 : cand0);
    const f4 o2 = (t0 == 0u) ? cand2 : ((t0 == 1u) ? cand0 : cand1);

    // 3 x 16B coalesced non-temporal streaming stores.
    __builtin_nontemporal_store(o0, out + idx);
    __builtin_nontemporal_store(o1, out + n4 + idx);
    __builtin_nontemporal_store(o2, out + 2u * n4 + idx);
  }
}

extern "C" void kernel_launch(void* const* d_in, const int* in_sizes, int n_in,
                              void* d_out, int out_size, void* d_ws,
                              size_t ws_size, hipStream_t stream) {
  (void)n_in; (void)out_size; (void)d_ws; (void)ws_size;
  const f4* x0 = (const f4*)d_in[0];
  const f4* x1 = (const f4*)d_in[1];
  const f4* x2 = (const f4*)d_in[2];
  f4* out = (f4*)d_out;

  const unsigned n4 = (unsigned)(in_sizes[0] / 4);  // 8,388,608 float4s/input

  // 4096 blocks x 256 threads (8 wave32 per block) -> 32K waves in flight,
  // 8 grid-stride iterations per thread so the prefetch pipeline has depth.
  dim3 block(256);
  dim3 grid(4096);
  pfe_kernel<<<grid, block, 0, stream>>>(x0, x1, x2, out, n4);
}